// TernaryLoss_76141180224172
// MI455X (gfx1250) — compile-verified
//
#include <hip/hip_runtime.h>
#include <stdint.h>

// ---------------------------------------------------------------------------
// TernaryLoss forward for MI455X (gfx1250, wave32).
//  - TDM (tensor_load_to_lds) stages the 3-channel halo tile into LDS
//    (one DMA per image per block, tracked with TENSORcnt).
//  - VALU/trans-bound census stencil, register-blocked vertical strips,
//    dx loop fully unrolled to minimize issue-slot overhead.
//  - Deterministic two-stage f64 reduction (no atomics).
// ---------------------------------------------------------------------------

#ifndef USE_TDM
#define USE_TDM 1
#endif

typedef uint32_t u32;
typedef u32 u32x4 __attribute__((ext_vector_type(4)));
typedef u32 u32x8 __attribute__((ext_vector_type(8)));

constexpr int Bn = 8, Cn = 3, Hn = 320, Wn = 1024;
constexpr int HWn = Hn * Wn;                 // 327680
constexpr int TH = 16, TW = 64;              // output tile
constexpr int HALO_H = TH + 6;               // 22
constexpr int HALO_W = TW + 6;               // 70
constexpr int RAW_N  = HALO_H * HALO_W;      // 1540
constexpr int IST    = HALO_W + 1;           // 71 (bank-friendly stride)
constexpr int GX = Wn / TW;                  // 16
constexpr int GY = Hn / TH;                  // 20
constexpr int NBLK = GX * GY * Bn;           // 2560

#if defined(__has_builtin)
# if __has_builtin(__builtin_amdgcn_s_wait_tensorcnt)
#  define WAIT_TENSOR0() __builtin_amdgcn_s_wait_tensorcnt(0)
# endif
#endif
#ifndef WAIT_TENSOR0
# define WAIT_TENSOR0() asm volatile("s_wait_tensorcnt 0x0" ::: "memory")
#endif

#if USE_TDM
// Issue one TDM load: 3D tile (dim0=HALO_W, dim1=HALO_H, dim2=Cn) of f32
// from global (row stride W, channel stride H*W) into contiguous LDS.
__device__ __forceinline__ void tdm_issue(u32 lds_off, const void* gptr) {
  const uint64_t ga = (uint64_t)(uintptr_t)gptr;
  u32x4 g0;
  g0[0] = 1u;                                          // count=1, user mode
  g0[1] = lds_off;                                     // LDS byte address
  g0[2] = (u32)ga;                                     // global addr [31:0]
  g0[3] = ((u32)(ga >> 32) & 0x01FFFFFFu) | (2u << 30); // addr[56:32] | type=2
  u32x8 g1;
  g1[0] = 2u << 16;                                    // data_size = 4B
  g1[1] = ((u32)Wn & 0xFFFFu) << 16;                   // tensor_dim0 lo16
  g1[2] = ((u32)Wn >> 16) | (((u32)Hn & 0xFFFFu) << 16); // dim0 hi | dim1 lo
  g1[3] = ((u32)Hn >> 16) | ((u32)HALO_W << 16);       // dim1 hi | tile_dim0
  g1[4] = (u32)HALO_H | (3u << 16);                    // tile_dim1 | tile_dim2
  g1[5] = (u32)Wn;                                     // dim0_stride [31:0]
  g1[6] = ((u32)HWn & 0xFFFFu) << 16;                  // stride0 hi=0 | stride1 lo16
  g1[7] = (u32)HWn >> 16;                              // stride1 [47:16]
  u32x4 g2;
  g2[0] = 3u;                                          // tensor_dim2
  g2[1] = 1u;                                          // tensor_dim3 (unused)
  g2[2] = 0u;                                          // dim2_stride lo
  g2[3] = 0u;                                          // stride hi | tile_dim3=0
  u32x4 g3 = {0u, 0u, 0u, 0u};
  asm volatile("tensor_load_to_lds %0, %1, %2, %3"
               :: "s"(g0), "s"(g1), "s"(g2), "s"(g3)
               : "memory");
}
#endif

__global__ __launch_bounds__(256) void ternary_census_kernel(
    const float* __restrict__ x, const float* __restrict__ y,
    double* __restrict__ partial)
{
#if USE_TDM
  __shared__ float rawX[Cn * RAW_N];
  __shared__ float rawY[Cn * RAW_N];
#endif
  __shared__ float  ixs[HALO_H * IST];
  __shared__ float  iys[HALO_H * IST];
  __shared__ double wred[8];

  const int tid = threadIdx.x;
  const int b  = blockIdx.z;
  const int h0 = blockIdx.y * TH;
  const int w0 = blockIdx.x * TW;
  // halo window clamped inside the image (border pixels are discarded anyway)
  int hs = h0 - 3; if (hs < 0) hs = 0; if (hs > Hn - HALO_H) hs = Hn - HALO_H;
  int ws = w0 - 3; if (ws < 0) ws = 0; if (ws > Wn - HALO_W) ws = Wn - HALO_W;
  const size_t base = (size_t)b * Cn * HWn + (size_t)hs * Wn + (size_t)ws;

#if USE_TDM
  if (tid < 32) {                         // one wave issues both DMAs
    tdm_issue((u32)(size_t)&rawX[0], (const void*)(x + base));
    tdm_issue((u32)(size_t)&rawY[0], (const void*)(y + base));
    WAIT_TENSOR0();
  }
  __syncthreads();
  for (int e = tid; e < RAW_N; e += 256) {
    const int r = e / HALO_W, c = e - r * HALO_W;
    ixs[r * IST + c] = (rawX[e] + rawX[e + RAW_N] + rawX[e + 2 * RAW_N]) * (1.0f / 3.0f);
    iys[r * IST + c] = (rawY[e] + rawY[e + RAW_N] + rawY[e + 2 * RAW_N]) * (1.0f / 3.0f);
  }
#else
  for (int e = tid; e < RAW_N; e += 256) {
    const int r = e / HALO_W, c = e - r * HALO_W;
    const size_t g = base + (size_t)r * Wn + (size_t)c;
    ixs[r * IST + c] = (x[g] + x[g + HWn] + x[g + 2 * HWn]) * (1.0f / 3.0f);
    iys[r * IST + c] = (y[g] + y[g + HWn] + y[g + 2 * HWn]) * (1.0f / 3.0f);
  }
#endif
  __syncthreads();

  // Each thread: 4-row vertical strip at one column of the tile.
  const int wl    = tid & (TW - 1);       // 0..63
  const int hg    = tid >> 6;             // 0..3
  const int hbase = h0 + hg * 4;
  const int Rb    = hbase - hs;           // LDS row of strip row 0
  const int Cc    = (w0 + wl) - ws;       // LDS col of this pixel column

  float cX[4], cY[4], acc[4];
#pragma unroll
  for (int i = 0; i < 4; ++i) {
    cX[i]  = ixs[(Rb + i) * IST + Cc];
    cY[i]  = iys[(Rb + i) * IST + Cc];
    acc[i] = 0.0f;
  }

#pragma unroll
  for (int dx = -3; dx <= 3; ++dx) {
    float vX[10], vY[10];
    const int i0 = (Rb - 3) * IST + (Cc + dx);
#pragma unroll
    for (int j = 0; j < 10; ++j) {        // 10-row column covers 4 centers x 7 dy
      vX[j] = ixs[i0 + j * IST];
      vY[j] = iys[i0 + j * IST];
    }
#pragma unroll
    for (int i = 0; i < 4; ++i) {
#pragma unroll
      for (int dy = 0; dy < 7; ++dy) {
        const float dXv = vX[i + dy] - cX[i];
        const float dYv = vY[i + dy] - cY[i];
        const float nX = dXv * __builtin_amdgcn_rsqf(__builtin_fmaf(dXv, dXv, 0.81f));
        const float nY = dYv * __builtin_amdgcn_rsqf(__builtin_fmaf(dYv, dYv, 0.81f));
        const float t  = nX - nY;
        // t^2/(0.1+t^2) = 1 - 0.1*rcp(0.1+t^2); constant part folded at the end
        acc[i] += __builtin_amdgcn_rcpf(__builtin_fmaf(t, t, 0.1f));
      }
    }
  }

  const int  gw = w0 + wl;
  const bool wv = (gw >= 3) & (gw <= Wn - 4);
  float tsum = 0.0f;
#pragma unroll
  for (int i = 0; i < 4; ++i) {
    const int  gh = hbase + i;
    const bool v  = wv & (gh >= 3) & (gh <= Hn - 4);
    tsum += v ? acc[i] : 0.0f;
  }

  // wave32 shuffle tree, then f64 cross-wave combine (deterministic)
  for (int off = 16; off > 0; off >>= 1)
    tsum += __shfl_down(tsum, off, 32);
  if ((tid & 31) == 0) wred[tid >> 5] = (double)tsum;
  __syncthreads();
  if (tid == 0) {
    double s = 0.0;
#pragma unroll
    for (int k = 0; k < 8; ++k) s += wred[k];
    partial[(blockIdx.z * GY + blockIdx.y) * GX + blockIdx.x] = s;
  }
}

__global__ __launch_bounds__(256) void final_reduce_kernel(
    const double* __restrict__ partial, float* __restrict__ out)
{
  __shared__ double sh[256];
  const int tid = threadIdx.x;
  double s = 0.0;
  for (int i = tid; i < NBLK; i += 256) s += partial[i];
  sh[tid] = s;
  __syncthreads();
  for (int off = 128; off > 0; off >>= 1) {
    if (tid < off) sh[tid] += sh[tid + off];
    __syncthreads();
  }
  if (tid == 0) {
    const double S     = sh[0];
    const double npix  = (double)Bn * (double)(Hn - 6) * (double)(Wn - 6);
    const double denom = (double)Bn * (double)Hn * (double)Wn;
    out[0] = (float)((npix - S * (0.1 / 49.0)) / denom);
  }
}

extern "C" void kernel_launch(void* const* d_in, const int* in_sizes, int n_in,
                              void* d_out, int out_size, void* d_ws, size_t ws_size,
                              hipStream_t stream) {
  (void)in_sizes; (void)n_in; (void)out_size; (void)ws_size;
  const float* x = (const float*)d_in[0];
  const float* y = (const float*)d_in[1];
  float*  out     = (float*)d_out;
  double* partial = (double*)d_ws;   // NBLK * 8 bytes = 20 KB

  dim3 grid(GX, GY, Bn), block(256);
  ternary_census_kernel<<<grid, block, 0, stream>>>(x, y, partial);
  final_reduce_kernel<<<1, 256, 0, stream>>>(partial, out);
}